// OptFC1EarlyTorch_48679159333252
// MI455X (gfx1250) — compile-verified
//
#include <hip/hip_runtime.h>
#include <math.h>
#include <stdint.h>

typedef float v2f __attribute__((ext_vector_type(2)));
typedef float v8f __attribute__((ext_vector_type(8)));

#define BM 128
#define BN 128
#define BK 32          // == E, so the first chunk's accumulator == x_e @ w_e.T
#define LDSK 36        // BK + 4 pad: 16B-aligned rows, conflict-free frag reads
#define NTHREADS 512   // 16 waves (wave32), arranged 4x4
#define BUF_FLOATS ((BM + BN) * LDSK)   // one double-buffer slot (X tile + W tile)
#define BOUND_F (-2.053748910631823f)

// CDNA5 async global->LDS copy (ASYNCcnt-tracked); data bypasses VGPRs.
// vdst = LDS byte address (low 32 bits of a flat shared pointer), vaddr = 64-bit global addr.
__device__ __forceinline__ void async_copy_b128(uint32_t lds_byte, const float* gaddr) {
    asm volatile("global_load_async_to_lds_b128 %0, %1, off"
                 :: "v"(lds_byte), "v"(gaddr)
                 : "memory");
}
__device__ __forceinline__ void wait_async0() {
    asm volatile("s_wait_asynccnt 0" ::: "memory");
}

__global__ __launch_bounds__(NTHREADS)
void fc1_gemm_gate_kernel(const float* __restrict__ X,      // [M, K]
                          const float* __restrict__ Wt,     // [N, K]
                          const float* __restrict__ Bias,   // [N]
                          const float* __restrict__ statsW, // [1]
                          const float* __restrict__ statsB, // [1]
                          float* __restrict__ Y,            // [M, N]
                          int M, int N, int K)
{
    __shared__ float smem[2 * BUF_FLOATS];   // 73728 B of the 320 KB/WGP LDS

    const int tid  = threadIdx.x;
    const int lane = tid & 31;
    const int wav  = tid >> 5;   // 0..15
    const int wm   = wav >> 2;   // 0..3  (wave row)
    const int wn   = wav & 3;    // 0..3  (wave col)
    const int half = lane >> 4;  // 0/1  (K-half selector for A/B frags)
    const int lr   = lane & 15;  // row-in-tile for A/B frag loads, col for C/D

    const int rowBase = blockIdx.y * BM;
    const int colBase = blockIdx.x * BN;

    v8f acc[2][2];   // full-K accumulator
    v8f accE[2][2];  // snapshot after first BK(=E) chunk: x_e @ w_e.T
    v8f accS[2][2];  // (x_e^2) @ (w_e^2).T
#pragma unroll
    for (int i = 0; i < 2; ++i)
#pragma unroll
        for (int j = 0; j < 2; ++j) {
            v8f z = {};
            acc[i][j] = z; accE[i][j] = z; accS[i][j] = z;
        }

    // Each thread owns 2 b128 slots per matrix per chunk (512 thr * 2 * 16B = 16 KB tile).
    int ldr[2], ldc[2];
#pragma unroll
    for (int i = 0; i < 2; ++i) {
        const int idx = tid + i * NTHREADS;   // 1024 float4 slots = 128 rows x 8
        ldr[i] = idx >> 3;
        ldc[i] = (idx & 7) << 2;
    }

    auto issue_chunk = [&](int buf, int k0) {
#pragma unroll
        for (int i = 0; i < 2; ++i) {
            const uint32_t lx = (uint32_t)(uintptr_t)
                &smem[buf * BUF_FLOATS + ldr[i] * LDSK + ldc[i]];
            async_copy_b128(lx, X + (size_t)(rowBase + ldr[i]) * K + k0 + ldc[i]);
            const uint32_t lw = (uint32_t)(uintptr_t)
                &smem[buf * BUF_FLOATS + BM * LDSK + ldr[i] * LDSK + ldc[i]];
            async_copy_b128(lw, Wt + (size_t)(colBase + ldr[i]) * K + k0 + ldc[i]);
        }
    };

    // Prologue: fill buffer 0.
    issue_chunk(0, 0);
    wait_async0();
    __syncthreads();

    const int nChunks = K / BK;
    for (int ch = 0; ch < nChunks; ++ch) {
        const int cur = ch & 1;
        if (ch + 1 < nChunks) issue_chunk(cur ^ 1, (ch + 1) * BK);

        const float* Xs = &smem[cur * BUF_FLOATS];
        const float* Ws = Xs + BM * LDSK;
        const bool first = (ch == 0);

#pragma unroll
        for (int kk = 0; kk < BK; kk += 4) {
            v2f a[2], b[2];
#pragma unroll
            for (int t = 0; t < 2; ++t) {
                // A 16x4 layout: lane(0-15)=rows,K={0,1}; lane(16-31)=rows,K={2,3}
                a[t] = *(const v2f*)&Xs[(wm * 32 + t * 16 + lr) * LDSK + kk + 2 * half];
                // B 4x16 layout is the identical addressing pattern on W rows
                b[t] = *(const v2f*)&Ws[(wn * 32 + t * 16 + lr) * LDSK + kk + 2 * half];
            }
#pragma unroll
            for (int i = 0; i < 2; ++i)
#pragma unroll
                for (int j = 0; j < 2; ++j)
                    acc[i][j] = __builtin_amdgcn_wmma_f32_16x16x4_f32(
                        false, a[i], false, b[j], (short)0, acc[i][j], false, false);
            if (first) {
                v2f a2[2], b2[2];
#pragma unroll
                for (int t = 0; t < 2; ++t) { a2[t] = a[t] * a[t]; b2[t] = b[t] * b[t]; }
#pragma unroll
                for (int i = 0; i < 2; ++i)
#pragma unroll
                    for (int j = 0; j < 2; ++j)
                        accS[i][j] = __builtin_amdgcn_wmma_f32_16x16x4_f32(
                            false, a2[i], false, b2[j], (short)0, accS[i][j], false, false);
            }
        }

        if (first) {  // accumulator now holds exactly the K<E partial sum
#pragma unroll
            for (int i = 0; i < 2; ++i)
#pragma unroll
                for (int j = 0; j < 2; ++j) accE[i][j] = acc[i][j];
        }

        // Next buffer's async copies must have landed, and all waves must be
        // done reading the current buffer, before the next iteration.
        wait_async0();
        __syncthreads();
    }

    // Epilogue: gate + bias + store.
    const float swv = statsW[0];
    const float sbv = statsB[0];
    const float invE = 1.0f / 32.0f;

#pragma unroll
    for (int i = 0; i < 2; ++i) {
#pragma unroll
        for (int j = 0; j < 2; ++j) {
            const int gm0 = rowBase + wm * 32 + i * 16 + 8 * half; // + r below
            const int gn  = colBase + wn * 32 + j * 16 + lr;
            const float bn = Bias[gn];
#pragma unroll
            for (int r = 0; r < 8; ++r) {
                const float y1   = accE[i][j][r] + bn;
                const float csq  = accS[i][j][r];
                const float mean = y1 * invE * swv + sbv;
                const float var  = (csq * invE - mean * mean) * swv * swv;
                const float ts   = mean / sqrtf(var * invE);  // NaN compares false
                const float tot  = acc[i][j][r] + bn;
                Y[(size_t)(gm0 + r) * N + gn] = (ts < BOUND_F) ? 0.0f : tot;
            }
        }
    }
}

extern "C" void kernel_launch(void* const* d_in, const int* in_sizes, int n_in,
                              void* d_out, int out_size, void* d_ws, size_t ws_size,
                              hipStream_t stream) {
    const float* X  = (const float*)d_in[0];
    const float* Wt = (const float*)d_in[1];
    const float* Bi = (const float*)d_in[2];
    const float* sw = (const float*)d_in[3];
    const float* sb = (const float*)d_in[4];
    float* Y = (float*)d_out;

    const int N = in_sizes[2];               // OUT = 8192
    const int K = in_sizes[1] / N;           // H   = 2048
    const int M = in_sizes[0] / K;           // B*S = 4096

    dim3 grid(N / BN, M / BM);
    fc1_gemm_gate_kernel<<<grid, NTHREADS, 0, stream>>>(X, Wt, Bi, sw, sb, Y, M, N, K);
}